// LGN_38285338476612
// MI455X (gfx1250) — compile-verified
//
#include <hip/hip_runtime.h>
#include <hip/hip_bf16.h>

#define N_USERS 100000
#define N_ITEMS 50000
#define N_NODES (N_USERS + N_ITEMS)
#define EMB_DIM 64
#define BATCH   4096

typedef float v2f __attribute__((ext_vector_type(2)));
typedef float v8f __attribute__((ext_vector_type(8)));

// ---------------- zero helpers ----------------
__global__ __launch_bounds__(256) void lgn_zero4(float4* __restrict__ p, int n4) {
    int i = blockIdx.x * 256 + threadIdx.x;
    if (i < n4) p[i] = make_float4(0.f, 0.f, 0.f, 0.f);
}

__global__ void lgn_zero_scal(float* scal) {
    scal[0] = 0.f;   // mf accumulator (sum of maxi)
    scal[1] = 0.f;   // reg accumulator (sum of squares)
}

// ---------------- SpMM: y[r] += v * x[c]  (COO, atomic accumulate) ----------------
// 16 threads per edge, each handles one float4 chunk of the 64-dim row.
__global__ __launch_bounds__(256) void lgn_spmm(
    const float* __restrict__ x,          // dense input  [N_NODES,64] (ignored if first)
    const float* __restrict__ uemb,       // used only when first != 0
    const float* __restrict__ iemb,
    const int*   __restrict__ rows,
    const int*   __restrict__ cols,
    const float* __restrict__ vals,
    float*       __restrict__ y,          // dense output [N_NODES,64], pre-zeroed
    int nEdges, int first)
{
    long t  = (long)blockIdx.x * 256 + threadIdx.x;
    int  e  = (int)(t >> 4);
    int  c4 = (int)(t & 15);
    if (e >= nEdges) return;

    int   col = cols[e];
    float v   = vals[e];
    const float* src;
    if (first) {
        src = (col < N_USERS) ? (uemb + (long)col * EMB_DIM)
                              : (iemb + (long)(col - N_USERS) * EMB_DIM);
    } else {
        src = x + (long)col * EMB_DIM;
    }
    float4 d = *(const float4*)(src + c4 * 4);

    float* dst = y + (long)rows[e] * EMB_DIM + c4 * 4;
    atomicAdd(dst + 0, v * d.x);
    atomicAdd(dst + 1, v * d.y);
    atomicAdd(dst + 2, v * d.z);
    atomicAdd(dst + 3, v * d.w);
}

// ---------------- BPR score differences via f32 WMMA ----------------
// Per block (one wave32): 16 batch rows. C = U(16x64) x (P-N)^T(64x16) via
// chained V_WMMA_F32_16X16X4_F32; diag(C) = pos_score - neg_score.
__global__ __launch_bounds__(32) void lgn_bpr_wmma(
    const float* __restrict__ emb,        // final all_emb [N_NODES,64]
    const int* __restrict__ users,
    const int* __restrict__ pos,
    const int* __restrict__ neg,
    float* __restrict__ acc_mf)
{
    const int l   = threadIdx.x;          // 0..31
    const int row = l & 15;               // M (and N) index this lane loads
    const int kh  = l >> 4;               // K half-select
    const int b   = blockIdx.x * 16 + row;

    const float* u = emb + (long)users[b] * EMB_DIM;
    const float* p = emb + (long)(N_USERS + pos[b]) * EMB_DIM;
    const float* n = emb + (long)(N_USERS + neg[b]) * EMB_DIM;

    v8f c = {};
#pragma unroll
    for (int k = 0; k < EMB_DIM; k += 4) {
        const int off = k + 2 * kh;                 // even -> 8B aligned
        float2 ua = *(const float2*)(u + off);
        float2 pa = *(const float2*)(p + off);
        float2 na = *(const float2*)(n + off);
        v2f a;  a.x  = ua.x;        a.y  = ua.y;
        v2f bb; bb.x = pa.x - na.x; bb.y = pa.y - na.y;
        // (neg_a, A, neg_b, B, c_mod, C, reuse_a, reuse_b)
        c = __builtin_amdgcn_wmma_f32_16x16x4_f32(
                false, a, false, bb, (short)0, c, false, false);
    }

    // Diagonal extraction:
    //   lanes 0..7  hold diag[i]   in c[i]      (i = lane)
    //   lanes 24..31 hold diag[i+8] in c[i]     (i = lane-24)
    const bool valid = (l < 8) || (l >= 24);
    const int  idx   = (l < 8) ? l : (l - 24);
    float d = c[0];
    d = (idx == 1) ? c[1] : d;
    d = (idx == 2) ? c[2] : d;
    d = (idx == 3) ? c[3] : d;
    d = (idx == 4) ? c[4] : d;
    d = (idx == 5) ? c[5] : d;
    d = (idx == 6) ? c[6] : d;
    d = (idx == 7) ? c[7] : d;

    if (valid) {
        float sg   = 1.0f / (1.0f + __expf(-d));    // sigmoid(pos - neg)
        float maxi = __logf(sg + 1e-10f);
        atomicAdd(acc_mf, maxi);
    }
}

// ---------------- L2 regularizer over initial embeddings ----------------
__global__ __launch_bounds__(256) void lgn_reg(
    const float* __restrict__ uemb, const float* __restrict__ iemb,
    const int* __restrict__ users, const int* __restrict__ pos,
    const int* __restrict__ neg, float* __restrict__ acc_reg)
{
    int i = blockIdx.x * 256 + threadIdx.x;
    if (i >= BATCH) return;
    const float4* u = (const float4*)(uemb + (long)users[i] * EMB_DIM);
    const float4* p = (const float4*)(iemb + (long)pos[i]  * EMB_DIM);
    const float4* n = (const float4*)(iemb + (long)neg[i]  * EMB_DIM);
    float s = 0.f;
#pragma unroll
    for (int k = 0; k < EMB_DIM / 4; ++k) {
        float4 a = u[k], b = p[k], d = n[k];
        s += a.x*a.x + a.y*a.y + a.z*a.z + a.w*a.w;
        s += b.x*b.x + b.y*b.y + b.z*b.z + b.w*b.w;
        s += d.x*d.x + d.y*d.y + d.z*d.z + d.w*d.w;
    }
    atomicAdd(acc_reg, s);
}

// ---------------- finalize scalars ----------------
__global__ void lgn_finalize(const float* __restrict__ scal, float* __restrict__ out) {
    out[0] = -scal[0] / (float)BATCH;                     // mf_loss
    out[1] = 1e-4f * 0.5f * scal[1] / (float)BATCH;       // reg_loss
}

extern "C" void kernel_launch(void* const* d_in, const int* in_sizes, int n_in,
                              void* d_out, int out_size, void* d_ws, size_t ws_size,
                              hipStream_t stream) {
    const float* uemb = (const float*)d_in[0];
    const float* iemb = (const float*)d_in[1];
    const float* vals = (const float*)d_in[2];
    const int*   rows = (const int*)d_in[3];
    const int*   cols = (const int*)d_in[4];
    const int*   users= (const int*)d_in[5];
    const int*   pos  = (const int*)d_in[6];
    const int*   neg  = (const int*)d_in[7];
    float* out = (float*)d_out;
    const int nE = in_sizes[2];

    char*  ws   = (char*)d_ws;
    float* scal = (float*)ws;                                 // 2 accumulators
    float* bufA = (float*)(ws + 256);
    float* bufB = bufA + (size_t)N_NODES * EMB_DIM;

    const int n4      = (N_NODES * EMB_DIM) / 4;              // 2,400,000
    const int zBlocks = (n4 + 255) / 256;
    const int sBlocks = (int)(((long)nE * 16 + 255) / 256);

    lgn_zero_scal<<<1, 1, 0, stream>>>(scal);

    // layer 1: embeddings -> A
    lgn_zero4<<<zBlocks, 256, 0, stream>>>((float4*)bufA, n4);
    lgn_spmm <<<sBlocks, 256, 0, stream>>>(nullptr, uemb, iemb, rows, cols, vals, bufA, nE, 1);
    // layer 2: A -> B
    lgn_zero4<<<zBlocks, 256, 0, stream>>>((float4*)bufB, n4);
    lgn_spmm <<<sBlocks, 256, 0, stream>>>(bufA, nullptr, nullptr, rows, cols, vals, bufB, nE, 0);
    // layer 3: B -> A
    lgn_zero4<<<zBlocks, 256, 0, stream>>>((float4*)bufA, n4);
    lgn_spmm <<<sBlocks, 256, 0, stream>>>(bufB, nullptr, nullptr, rows, cols, vals, bufA, nE, 0);

    // BPR scores via WMMA (256 groups of 16 batch rows, one wave each)
    lgn_bpr_wmma<<<BATCH / 16, 32, 0, stream>>>(bufA, users, pos, neg, &scal[0]);
    // regularizer
    lgn_reg<<<(BATCH + 255) / 256, 256, 0, stream>>>(uemb, iemb, users, pos, neg, &scal[1]);

    lgn_finalize<<<1, 1, 0, stream>>>(scal, out);
}